// net_55903294324835
// MI455X (gfx1250) — compile-verified
//
#include <hip/hip_runtime.h>
#include <hip/hip_bf16.h>

// Model dims (fixed by the reference)
#define DIM 2
#define NK 2
#define CUM 14
#define HID 128
#define OUTD 3

typedef __attribute__((ext_vector_type(2))) float v2f;
typedef __attribute__((ext_vector_type(4))) float v4f;
typedef __attribute__((ext_vector_type(8))) float v8f;

// ---------------------------------------------------------------------------
// Kernel 0: zero workspace accumulators
// ---------------------------------------------------------------------------
__global__ void k_zero(float* p, int n) {
    int i = blockIdx.x * blockDim.x + threadIdx.x;
    if (i < n) p[i] = 0.0f;
}

// ---------------------------------------------------------------------------
// Kernel 1: edge pass -> deg[dst] += 1 ; sumx[dst] += x[src]   (f32 L2 atomics)
// ---------------------------------------------------------------------------
__global__ void k_deg_sumx(const int* __restrict__ src, const int* __restrict__ dst,
                           const float2* __restrict__ x2,
                           float* __restrict__ deg, float* __restrict__ sumx, int E) {
    int e = blockIdx.x * blockDim.x + threadIdx.x;
    if (e >= E) return;
    int s = src[e], d = dst[e];
    float2 xs = x2[s];
    atomicAdd(&deg[d], 1.0f);
    atomicAdd(&sumx[2 * d + 0], xs.x);
    atomicAdd(&sumx[2 * d + 1], xs.y);
}

// ---------------------------------------------------------------------------
// Kernel 2: node pass -> xd = x + tau * (sumx/deg - x)
// ---------------------------------------------------------------------------
__global__ void k_diffuse(const float2* __restrict__ x2, const float2* __restrict__ sumx2,
                          const float* __restrict__ deg, const float* __restrict__ tau_p,
                          float2* __restrict__ xd2, int N) {
    int i = blockIdx.x * blockDim.x + threadIdx.x;
    if (i >= N) return;
    float dg = deg[i]; dg = dg < 1.0f ? 1.0f : dg;
    float inv = 1.0f / dg;
    float tau = *tau_p;
    float2 a = x2[i], sm = sumx2[i];
    float2 r;
    r.x = a.x + tau * (sm.x * inv - a.x);
    r.y = a.y + tau * (sm.y * inv - a.y);
    xd2[i] = r;
}

// ---------------------------------------------------------------------------
// Kernel 3: edge pass -> hop1 scatter (2ch -> 4ch), sheaf map R, and the
//           xd-part of the fused 14-ch aggregation (channels 0..1).
//   R = [xd[src], xd[dst]] @ W_sheaf + b_sheaf   (per edge, 4x4), NT stream.
// ---------------------------------------------------------------------------
__global__ void k_hop1_R(const int* __restrict__ src, const int* __restrict__ dst,
                         const float2* __restrict__ xd2, const float* __restrict__ K,
                         const float* __restrict__ Wsh, const float* __restrict__ bsh,
                         float* __restrict__ h1, float* __restrict__ agg,
                         float* __restrict__ R, int E, int r_aligned16) {
    int e = blockIdx.x * blockDim.x + threadIdx.x;
    if (e >= E) return;
    int s = src[e], d = dst[e];
    float2 xs = xd2[s];
    float2 xdd = xd2[d];
    float k0 = K[e], k1 = K[E + e];
    // hop1 scatter
    atomicAdd(&h1[4 * d + 0], k0 * xs.x);
    atomicAdd(&h1[4 * d + 1], k0 * xs.y);
    atomicAdd(&h1[4 * d + 2], k1 * xs.x);
    atomicAdd(&h1[4 * d + 3], k1 * xs.y);
    // fused agg channels 0..1 (out[0] = xd)
    atomicAdd(&agg[CUM * d + 0], xs.x);
    atomicAdd(&agg[CUM * d + 1], xs.y);
    // sheaf restriction map (write-once 51MB stream -> non-temporal)
    v4f r;
    r.x = bsh[0] + xs.x * Wsh[0]  + xs.y * Wsh[4]  + xdd.x * Wsh[8]  + xdd.y * Wsh[12];
    r.y = bsh[1] + xs.x * Wsh[1]  + xs.y * Wsh[5]  + xdd.x * Wsh[9]  + xdd.y * Wsh[13];
    r.z = bsh[2] + xs.x * Wsh[2]  + xs.y * Wsh[6]  + xdd.x * Wsh[10] + xdd.y * Wsh[14];
    r.w = bsh[3] + xs.x * Wsh[3]  + xs.y * Wsh[7]  + xdd.x * Wsh[11] + xdd.y * Wsh[15];
    if (r_aligned16) {
        __builtin_nontemporal_store(r, (v4f*)(R + 4 * (size_t)e));
    } else {
        __builtin_nontemporal_store(r.x, R + 4 * (size_t)e + 0);
        __builtin_nontemporal_store(r.y, R + 4 * (size_t)e + 1);
        __builtin_nontemporal_store(r.z, R + 4 * (size_t)e + 2);
        __builtin_nontemporal_store(r.w, R + 4 * (size_t)e + 3);
    }
}

// ---------------------------------------------------------------------------
// Kernel 4: edge pass -> hop2 scatter (4ch -> 8ch) and the h1-part of the
//           fused aggregation (channels 2..5).
// ---------------------------------------------------------------------------
__global__ void k_hop2(const int* __restrict__ src, const int* __restrict__ dst,
                       const float4* __restrict__ h14, const float* __restrict__ K,
                       float* __restrict__ h2, float* __restrict__ agg, int E) {
    int e = blockIdx.x * blockDim.x + threadIdx.x;
    if (e >= E) return;
    int s = src[e], d = dst[e];
    float k0 = K[e], k1 = K[E + e];
    float4 v = h14[s];
    atomicAdd(&h2[8 * d + 0], k0 * v.x);
    atomicAdd(&h2[8 * d + 1], k0 * v.y);
    atomicAdd(&h2[8 * d + 2], k0 * v.z);
    atomicAdd(&h2[8 * d + 3], k0 * v.w);
    atomicAdd(&h2[8 * d + 4], k1 * v.x);
    atomicAdd(&h2[8 * d + 5], k1 * v.y);
    atomicAdd(&h2[8 * d + 6], k1 * v.z);
    atomicAdd(&h2[8 * d + 7], k1 * v.w);
    atomicAdd(&agg[CUM * d + 2], v.x);
    atomicAdd(&agg[CUM * d + 3], v.y);
    atomicAdd(&agg[CUM * d + 4], v.z);
    atomicAdd(&agg[CUM * d + 5], v.w);
}

// ---------------------------------------------------------------------------
// Kernel 5: edge pass -> h2-part of the fused aggregation (channels 6..13)
// ---------------------------------------------------------------------------
__global__ void k_agg_h2(const int* __restrict__ src, const int* __restrict__ dst,
                         const float4* __restrict__ h24, float* __restrict__ agg, int E) {
    int e = blockIdx.x * blockDim.x + threadIdx.x;
    if (e >= E) return;
    int s = src[e], d = dst[e];
    float4 a = h24[2 * s + 0];
    float4 b = h24[2 * s + 1];
    atomicAdd(&agg[CUM * d + 6],  a.x);
    atomicAdd(&agg[CUM * d + 7],  a.y);
    atomicAdd(&agg[CUM * d + 8],  a.z);
    atomicAdd(&agg[CUM * d + 9],  a.w);
    atomicAdd(&agg[CUM * d + 10], b.x);
    atomicAdd(&agg[CUM * d + 11], b.y);
    atomicAdd(&agg[CUM * d + 12], b.z);
    atomicAdd(&agg[CUM * d + 13], b.w);
}

// ---------------------------------------------------------------------------
// Kernel 6: per-node MLP head with WMMA f32 16x16x4.
//   t   = relu((agg/deg) @ W_conv + b_conv)    (14x14, split across lane halves)
//   hdn = relu(t @ W1 + b1)                    (16x128 via V_WMMA_F32_16X16X4_F32)
//   y   = hdn @ W2 + b2                        (128x3, split halves + shfl combine)
// One wave handles a 16-node tile; 4 waves / block -> 64 nodes per block.
// ---------------------------------------------------------------------------
#define HSTR 17   // hdn LDS row stride (pad 16->17: conflict-free WMMA-epilogue stores)

__global__ __launch_bounds__(128) void k_mlp(
    const float* __restrict__ agg, const float* __restrict__ deg,
    const float* __restrict__ W_conv, const float* __restrict__ b_conv,
    const float* __restrict__ W1, const float* __restrict__ b1,
    const float* __restrict__ W2, const float* __restrict__ b2,
    float* __restrict__ y, int N) {
    // Per-wave staging.  T: 16(K-padded) x 16(M), stored T[k*16+m] (transposed).
    // H: hdn transposed H[col*HSTR+row].
    __shared__ float lds_t[4][16 * 16];
    __shared__ float lds_h[4][HID * HSTR];

    const int wave = threadIdx.x >> 5;
    const int lane = threadIdx.x & 31;
    const int half = lane >> 4;                 // 0 or 1
    const int mrow = lane & 15;
    const int nodeBase = (blockIdx.x * 4 + wave) * 16;
    float* Tt = lds_t[wave];
    float* Hh = lds_h[wave];

    // ---- stage 1: 14x14 linear + relu; lane half h computes rows h*7..h*7+6
    //      plus padding row 14+h.  Node = nodeBase + mrow (both halves). ----
    {
        int node = nodeBase + mrow;
        const int j0 = half * 7;
        if (node < N) {
            float dg = deg[node]; dg = dg < 1.0f ? 1.0f : dg;
            float inv = 1.0f / dg;
            float a[CUM];
#pragma unroll
            for (int c = 0; c < CUM; ++c) a[c] = agg[node * CUM + c] * inv;
#pragma unroll
            for (int jj = 0; jj < 7; ++jj) {
                int j = j0 + jj;
                float s = b_conv[j];
#pragma unroll
                for (int c = 0; c < CUM; ++c) s += a[c] * W_conv[c * CUM + j];
                Tt[j * 16 + mrow] = s > 0.0f ? s : 0.0f;
            }
            Tt[(14 + half) * 16 + mrow] = 0.0f;  // K padding 14 -> 16
        } else {
#pragma unroll
            for (int jj = 0; jj < 7; ++jj) Tt[(j0 + jj) * 16 + mrow] = 0.0f;
            Tt[(14 + half) * 16 + mrow] = 0.0f;
        }
    }
    __syncthreads();

    // ---- stage 2: hdn = relu(T @ W1 + b1) via V_WMMA_F32_16X16X4_F32 ----
    // ISA 16x4 f32 A layout: lanes 0-15 hold M=lane, VGPR0=K0 VGPR1=K1;
    // lanes 16-31 hold M=lane-16, VGPR0=K2 VGPR1=K3.  B (4x16 KxN) symmetric.
    const int khalf = half << 1;               // 0 or 2
#pragma unroll
    for (int nt = 0; nt < HID / 16; ++nt) {
        v8f acc = {};
        const int col = nt * 16 + mrow;
#pragma unroll
        for (int k0 = 0; k0 < 16; k0 += 4) {
            const int kb = k0 + khalf;
            // clamp address for padded K rows (A there is zero, product is zero)
            const int kr0 = kb     < CUM ? kb     : CUM - 1;
            const int kr1 = kb + 1 < CUM ? kb + 1 : CUM - 1;
            v2f a, b;
            a.x = Tt[(kb + 0) * 16 + mrow];
            a.y = Tt[(kb + 1) * 16 + mrow];
            b.x = W1[kr0 * HID + col];
            b.y = W1[kr1 * HID + col];
            acc = __builtin_amdgcn_wmma_f32_16x16x4_f32(
                false, a, false, b, (short)0, acc, false, false);
        }
        const float bb = b1[col];
#pragma unroll
        for (int i = 0; i < 8; ++i) {
            const int row = i + (half << 3);    // D: VGPR i = M i (lo) / i+8 (hi)
            float v = acc[i] + bb;
            Hh[col * HSTR + row] = v > 0.0f ? v : 0.0f;
        }
    }
    __syncthreads();

    // ---- stage 3: y = hdn @ W2 + b2; lane half h reduces c = h*64..h*64+63,
    //      then combine halves with a cross-lane shuffle. ----
    {
        int node = nodeBase + mrow;
        float s0 = 0.0f, s1 = 0.0f, s2 = 0.0f;
        const int c0 = half * (HID / 2);
#pragma unroll 4
        for (int cc = 0; cc < HID / 2; ++cc) {
            int c = c0 + cc;
            float h = Hh[c * HSTR + mrow];
            s0 += h * W2[c * 3 + 0];
            s1 += h * W2[c * 3 + 1];
            s2 += h * W2[c * 3 + 2];
        }
        s0 += __shfl_down(s0, 16, 32);
        s1 += __shfl_down(s1, 16, 32);
        s2 += __shfl_down(s2, 16, 32);
        if (half == 0 && node < N) {
            y[node * 3 + 0] = s0 + b2[0];
            y[node * 3 + 1] = s1 + b2[1];
            y[node * 3 + 2] = s2 + b2[2];
        }
    }
}

// ---------------------------------------------------------------------------
// Host-side launcher (graph-capture safe: launches only, workspace from d_ws)
// ---------------------------------------------------------------------------
extern "C" void kernel_launch(void* const* d_in, const int* in_sizes, int n_in,
                              void* d_out, int out_size, void* d_ws, size_t ws_size,
                              hipStream_t stream) {
    const float* x      = (const float*)d_in[0];
    const float* K      = (const float*)d_in[1];
    const float* tau    = (const float*)d_in[2];
    const float* W_sheaf= (const float*)d_in[3];
    const float* b_sheaf= (const float*)d_in[4];
    const float* W_conv = (const float*)d_in[5];
    const float* b_conv = (const float*)d_in[6];
    const float* W1     = (const float*)d_in[7];
    const float* b1     = (const float*)d_in[8];
    const float* W2     = (const float*)d_in[9];
    const float* b2     = (const float*)d_in[10];
    const int*   eidx   = (const int*)d_in[11];

    const int N = in_sizes[0] / DIM;
    const int E = in_sizes[11] / 2;
    const int* src = eidx;
    const int* dst = eidx + E;

    // Workspace layout (floats), every region 16B-aligned via NP padding:
    // [deg NP][sumx 2NP][h1 4NP][h2 8NP][agg 14NP][xd 2NP]
    const int NP = (N + 3) & ~3;
    float* ws   = (float*)d_ws;
    float* deg  = ws;
    float* sumx = deg  + NP;
    float* h1   = sumx + 2 * NP;
    float* h2   = h1   + 4 * NP;
    float* agg  = h2   + 8 * NP;
    float* xd   = agg  + 14 * NP;
    const int n_zero = 29 * NP;                 // accumulators needing zeros

    // Outputs: y [N*3] then R [E*4]
    float* y = (float*)d_out;
    float* R = y + (size_t)N * OUTD;
    const int r_aligned16 = (((uintptr_t)R & 15u) == 0) ? 1 : 0;

    const int TB = 256;
    const int gbE = (E + TB - 1) / TB;
    const int gbN = (N + TB - 1) / TB;

    k_zero<<<(n_zero + TB - 1) / TB, TB, 0, stream>>>(ws, n_zero);
    k_deg_sumx<<<gbE, TB, 0, stream>>>(src, dst, (const float2*)x, deg, sumx, E);
    k_diffuse<<<gbN, TB, 0, stream>>>((const float2*)x, (const float2*)sumx, deg,
                                      tau, (float2*)xd, N);
    k_hop1_R<<<gbE, TB, 0, stream>>>(src, dst, (const float2*)xd, K,
                                     W_sheaf, b_sheaf, h1, agg, R, E, r_aligned16);
    k_hop2<<<gbE, TB, 0, stream>>>(src, dst, (const float4*)h1, K, h2, agg, E);
    k_agg_h2<<<gbE, TB, 0, stream>>>(src, dst, (const float4*)h2, agg, E);
    k_mlp<<<(N + 63) / 64, 128, 0, stream>>>(agg, deg, W_conv, b_conv,
                                             W1, b1, W2, b2, y, N);
}